// HyperConv_17403207483834
// MI455X (gfx1250) — compile-verified
//
#include <hip/hip_runtime.h>
#include <stdint.h>

// HyperConv: 3 residual SpMM layers over 2 embedding tables.
// Strategy: build CSR once (hist + scan + scatter), then wave-per-row gather
// SpMM. Embedding table (51.2MB) is L2-resident on MI455X (192MB L2), so the
// random per-edge gathers hit L2. Edge lists are staged into LDS with the
// CDNA5 async global->LDS path (global_load_async_to_lds_b32 / s_wait_asynccnt),
// batched 128 edges per wait (one wait per average row), and broadcast-read
// with ds_load_b64.

#define EMB 128
#define WAVES_PER_BLOCK 8
#define THREADS 256
#define EDGE_BATCH 128  // edges staged per async batch (4 per lane)

__global__ void zero_kernel(int* __restrict__ p, int n) {
  int i = blockIdx.x * blockDim.x + threadIdx.x;
  if (i < n) p[i] = 0;
}

__global__ void hist_kernel(const int* __restrict__ rows, int* __restrict__ cnt, int nnz) {
  int i = blockIdx.x * blockDim.x + threadIdx.x;
  if (i < nnz) atomicAdd(&cnt[rows[i]], 1);
}

// Single-block scan over n counts -> exclusive prefix (row_start), row_start[n] = total.
__global__ __launch_bounds__(1024) void scan_kernel(const int* __restrict__ cnt,
                                                    int* __restrict__ row_start, int n) {
  __shared__ int sh[1024];
  __shared__ int carry_s;
  if (threadIdx.x == 0) carry_s = 0;
  __syncthreads();
  for (int base = 0; base < n; base += 1024) {
    int i = base + threadIdx.x;
    int v = (i < n) ? cnt[i] : 0;
    sh[threadIdx.x] = v;
    __syncthreads();
    for (int off = 1; off < 1024; off <<= 1) {
      int t = (threadIdx.x >= off) ? sh[threadIdx.x - off] : 0;
      __syncthreads();
      sh[threadIdx.x] += t;
      __syncthreads();
    }
    if (i < n) row_start[i] = carry_s + sh[threadIdx.x] - v;  // exclusive
    __syncthreads();
    if (threadIdx.x == 0) carry_s += sh[1023];
    __syncthreads();
  }
  if (threadIdx.x == 0) row_start[n] = carry_s;
}

__global__ void copy_kernel(const int* __restrict__ src, int* __restrict__ dst, int n) {
  int i = blockIdx.x * blockDim.x + threadIdx.x;
  if (i < n) dst[i] = src[i];
}

__global__ void scatter_kernel(const int* __restrict__ rows, const int* __restrict__ cols,
                               const float* __restrict__ vals, int* __restrict__ cursor,
                               int* __restrict__ csr_col, float* __restrict__ csr_val, int nnz) {
  int i = blockIdx.x * blockDim.x + threadIdx.x;
  if (i < nnz) {
    int r = rows[i];
    int pos = atomicAdd(&cursor[r], 1);
    csr_col[pos] = cols[i];
    csr_val[pos] = vals[i];
  }
}

// out[r] = in[r] + sum_{e in row r} val[e] * in[col[e]]
// One wave32 per row; lane l owns features [4l, 4l+4) as a float4.
__global__ __launch_bounds__(THREADS) void spmm_kernel(
    const float* __restrict__ in, float* __restrict__ out,
    const int* __restrict__ row_start,
    const int* __restrict__ csr_col, const float* __restrict__ csr_val,
    int n_node) {
  // Per-wave staging buffer: EDGE_BATCH interleaved (col:int32, val:f32) pairs.
  __shared__ unsigned long long ebuf[WAVES_PER_BLOCK][EDGE_BATCH];

  const int wave = threadIdx.x >> 5;
  const int lane = threadIdx.x & 31;
  const int row  = blockIdx.x * WAVES_PER_BLOCK + wave;
  if (row >= n_node) return;

  const int start = row_start[row];
  const int end   = row_start[row + 1];

  const float4* __restrict__ inv = reinterpret_cast<const float4*>(in);
  float4 acc = inv[(size_t)row * (EMB / 4) + lane];  // residual term

  // LDS byte offset of this wave's buffer (flat LDS address low 32 bits == DS offset).
  const unsigned lds_base = (unsigned)(uintptr_t)(&ebuf[wave][0]);

  for (int e0 = start; e0 < end; e0 += EDGE_BATCH) {
    const int nb = min(EDGE_BATCH, end - e0);

    // Stage nb edges into LDS: lane l handles edges l, l+32, l+64, l+96.
    // Up to 8 async instructions in flight before a single wait.
#pragma unroll
    for (int s = 0; s < EDGE_BATCH / 32; ++s) {
      const int k = s * 32 + lane;
      if (k < nb) {
        const unsigned ldsc = lds_base + 8u * (unsigned)k;  // col slot
        const unsigned ldsv = ldsc + 4u;                    // val slot
        const unsigned goff = (unsigned)(e0 + k) * 4u;      // byte offset in both arrays
        asm volatile(
            "global_load_async_to_lds_b32 %0, %2, %3\n\t"
            "global_load_async_to_lds_b32 %1, %2, %4"
            :
            : "v"(ldsc), "v"(ldsv), "v"(goff), "s"(csr_col), "s"(csr_val)
            : "memory");
      }
    }
    asm volatile("s_wait_asynccnt 0" ::: "memory");

    int j = 0;
    // Unroll x8: 8 independent L2 gathers in flight per wave.
    for (; j + 8 <= nb; j += 8) {
      float  vv[8];
      float4 bb[8];
#pragma unroll
      for (int u = 0; u < 8; ++u) {
        unsigned long long cv = ebuf[wave][j + u];   // ds_load_b64, broadcast
        unsigned c = (unsigned)(cv & 0xffffffffu);
        vv[u] = __uint_as_float((unsigned)(cv >> 32));
        bb[u] = inv[(size_t)c * (EMB / 4) + lane];   // coalesced 512B/wave, L2-resident
      }
#pragma unroll
      for (int u = 0; u < 8; ++u) {
        acc.x = fmaf(vv[u], bb[u].x, acc.x);
        acc.y = fmaf(vv[u], bb[u].y, acc.y);
        acc.z = fmaf(vv[u], bb[u].z, acc.z);
        acc.w = fmaf(vv[u], bb[u].w, acc.w);
      }
    }
    for (; j < nb; ++j) {
      unsigned long long cv = ebuf[wave][j];
      unsigned c = (unsigned)(cv & 0xffffffffu);
      float v = __uint_as_float((unsigned)(cv >> 32));
      float4 b = inv[(size_t)c * (EMB / 4) + lane];
      acc.x = fmaf(v, b.x, acc.x);
      acc.y = fmaf(v, b.y, acc.y);
      acc.z = fmaf(v, b.z, acc.z);
      acc.w = fmaf(v, b.w, acc.w);
    }
  }

  reinterpret_cast<float4*>(out)[(size_t)row * (EMB / 4) + lane] = acc;
}

extern "C" void kernel_launch(void* const* d_in, const int* in_sizes, int n_in,
                              void* d_out, int out_size, void* d_ws, size_t ws_size,
                              hipStream_t stream) {
  (void)n_in; (void)out_size; (void)ws_size;
  const int*   rows = (const int*)d_in[0];
  const int*   cols = (const int*)d_in[1];
  const float* vals = (const float*)d_in[2];
  const float* emb  = (const float*)d_in[3];
  const float* mo   = (const float*)d_in[4];
  const int nnz    = in_sizes[0];
  const int n_node = in_sizes[3] / EMB;

  // Workspace layout (~78 MB): csr_col | csr_val | row_start | cursor | S (scratch emb)
  char* w = (char*)d_ws;
  int*   csr_col   = (int*)w;    w += (size_t)nnz * sizeof(int);
  float* csr_val   = (float*)w;  w += (size_t)nnz * sizeof(float);
  int*   row_start = (int*)w;    w += (size_t)(n_node + 1) * sizeof(int);
  int*   cursor    = (int*)w;    w += (size_t)n_node * sizeof(int);
  w = (char*)(((uintptr_t)w + 255) & ~(uintptr_t)255);
  float* S = (float*)w;

  float* out_id = (float*)d_out;
  float* out_mo = out_id + (size_t)n_node * EMB;

  const int tb = 256;
  // CSR build (counts live in `cursor` first, then scanned into row_start).
  zero_kernel<<<(n_node + tb - 1) / tb, tb, 0, stream>>>(cursor, n_node);
  hist_kernel<<<(nnz + tb - 1) / tb, tb, 0, stream>>>(rows, cursor, nnz);
  scan_kernel<<<1, 1024, 0, stream>>>(cursor, row_start, n_node);
  copy_kernel<<<(n_node + tb - 1) / tb, tb, 0, stream>>>(row_start, cursor, n_node);
  scatter_kernel<<<(nnz + tb - 1) / tb, tb, 0, stream>>>(rows, cols, vals, cursor,
                                                         csr_col, csr_val, nnz);

  const int grid = (n_node + WAVES_PER_BLOCK - 1) / WAVES_PER_BLOCK;
  // id chain: emb -> out_id -> S -> out_id
  spmm_kernel<<<grid, THREADS, 0, stream>>>(emb,    out_id, row_start, csr_col, csr_val, n_node);
  spmm_kernel<<<grid, THREADS, 0, stream>>>(out_id, S,      row_start, csr_col, csr_val, n_node);
  spmm_kernel<<<grid, THREADS, 0, stream>>>(S,      out_id, row_start, csr_col, csr_val, n_node);
  // mo chain: mo -> out_mo -> S -> out_mo
  spmm_kernel<<<grid, THREADS, 0, stream>>>(mo,     out_mo, row_start, csr_col, csr_val, n_node);
  spmm_kernel<<<grid, THREADS, 0, stream>>>(out_mo, S,      row_start, csr_col, csr_val, n_node);
  spmm_kernel<<<grid, THREADS, 0, stream>>>(S,      out_mo, row_start, csr_col, csr_val, n_node);
}